// MultiViewRecommender_17789754540971
// MI455X (gfx1250) — compile-verified
//
#include <hip/hip_runtime.h>
#include <hip/hip_bf16.h>
#include <math.h>

// MI455X / gfx1250, wave32. One wave owns a 16-row tile; all row-wise
// reductions (dots, norms, attention projection, gathered negatives) are
// computed as diagonals of 16x16 f32 WMMA tiles (V_WMMA_F32_16X16X4_F32),
// preserving full f32 accuracy. Memory-bound workload (~0.9 GB traffic).

typedef __attribute__((ext_vector_type(2))) float v2f;
typedef __attribute__((ext_vector_type(4))) float v4f;
typedef __attribute__((ext_vector_type(8))) float v8f;

#define WMMA_F32(A, B, C) \
    __builtin_amdgcn_wmma_f32_16x16x4_f32(false, (A), false, (B), (short)0, (C), false, false)

__device__ __forceinline__ float vsel8(v8f a, int i) {
    float r = a[0];
    r = (i == 1) ? a[1] : r;
    r = (i == 2) ? a[2] : r;
    r = (i == 3) ? a[3] : r;
    r = (i == 4) ? a[4] : r;
    r = (i == 5) ? a[5] : r;
    r = (i == 6) ? a[6] : r;
    r = (i == 7) ? a[7] : r;
    return r;
}

__device__ __forceinline__ float softplus_f(float x) {
    // numerically stable log(1+exp(x))
    return (x > 0.f) ? (x + log1pf(expf(-x))) : log1pf(expf(x));
}

__global__ void zero_scalar_kernel(float* p) {
    if (threadIdx.x == 0 && blockIdx.x == 0) *p = 0.f;
}

__global__ __launch_bounds__(128) void mvrec_wmma_kernel(
    const float* __restrict__ g1, const float* __restrict__ g2,
    const float* __restrict__ s1, const float* __restrict__ s2,
    const float* __restrict__ attw,
    const int* __restrict__ neg_g_idx, const int* __restrict__ neg_s_idx,
    const int* __restrict__ neg_c_idx,
    float* __restrict__ out, float* __restrict__ loss,
    int n_items, int n_tiles)
{
    __shared__ float red[4][16][16];   // [wave][quantity][row]

    const int lane = threadIdx.x & 31;
    const int wave = threadIdx.x >> 5;
    const int tile = blockIdx.x * 4 + wave;
    if (tile >= n_tiles) return;       // wave-uniform guard (never splits a wave)

    const int r0    = tile * 16;
    const int mrow  = lane & 15;       // matrix row (A) / column (B) this lane feeds
    const int row   = r0 + mrow;
    const int khalf = (lane >> 4) * 2; // K sub-offset per lane half (A/B layout)

    const int j1 = neg_g_idx[row];
    const int j2 = neg_s_idx[row];
    const int j3 = neg_c_idx[row];

    const int off_own = row * 128 + khalf;
    const int off_g2n = j1 * 128 + khalf;
    const int off_s2n = j2 * 128 + khalf;
    const int off_cn  = j3 * 128 + khalf;

    // 16 accumulators: diag(C) = per-row reductions
    v8f acc_gg = {}, acc_ng1 = {}, acc_ng2 = {};
    v8f acc_ss = {}, acc_ns1 = {}, acc_ns2 = {};
    v8f acc_cc = {}, acc_nga = {}, acc_nsa = {}, acc_att = {};
    v8f acc_gn = {}, acc_ng2n = {}, acc_sn = {}, acc_ns2n = {};
    v8f acc_cn = {}, acc_nsan = {};

    for (int k0 = 0; k0 < 128; k0 += 4) {
        v2f a_g1  = *(const v2f*)(g1 + off_own + k0);
        v2f a_g2  = *(const v2f*)(g2 + off_own + k0);
        v2f a_s1  = *(const v2f*)(s1 + off_own + k0);
        v2f a_s2  = *(const v2f*)(s2 + off_own + k0);
        v2f b_g2n = *(const v2f*)(g2 + off_g2n + k0);
        v2f b_s2n = *(const v2f*)(s2 + off_s2n + k0);
        v2f c1    = *(const v2f*)(s1 + off_cn + k0);
        v2f c2    = *(const v2f*)(s2 + off_cn + k0);
        v2f w_g   = *(const v2f*)(attw + khalf + k0);
        v2f w_s   = *(const v2f*)(attw + 128 + khalf + k0);

        v2f a_ga  = 0.5f * (a_g1 + a_g2);
        v2f a_sa  = 0.5f * (a_s1 + a_s2);
        v2f b_sac = 0.5f * (c1 + c2);

        acc_gg   = WMMA_F32(a_g1, a_g2, acc_gg);    // g1 . g2
        acc_ng1  = WMMA_F32(a_g1, a_g1, acc_ng1);   // ||g1||^2
        acc_ng2  = WMMA_F32(a_g2, a_g2, acc_ng2);   // ||g2||^2
        acc_ss   = WMMA_F32(a_s1, a_s2, acc_ss);    // s1 . s2
        acc_ns1  = WMMA_F32(a_s1, a_s1, acc_ns1);
        acc_ns2  = WMMA_F32(a_s2, a_s2, acc_ns2);
        acc_cc   = WMMA_F32(a_ga, a_sa, acc_cc);    // ga . sa
        acc_nga  = WMMA_F32(a_ga, a_ga, acc_nga);
        acc_nsa  = WMMA_F32(a_sa, a_sa, acc_nsa);
        acc_att  = WMMA_F32(a_ga, w_g, acc_att);    // ga . w[:128]
        acc_att  = WMMA_F32(a_sa, w_s, acc_att);    // + sa . w[128:]
        acc_gn   = WMMA_F32(a_g1, b_g2n, acc_gn);   // g1[i] . g2[j1]
        acc_ng2n = WMMA_F32(b_g2n, b_g2n, acc_ng2n);
        acc_sn   = WMMA_F32(a_s1, b_s2n, acc_sn);   // s1[i] . s2[j2]
        acc_ns2n = WMMA_F32(b_s2n, b_s2n, acc_ns2n);
        acc_cn   = WMMA_F32(a_ga, b_sac, acc_cn);   // ga[i] . sa[j3]
        acc_nsan = WMMA_F32(b_sac, b_sac, acc_nsan);
    }

    // ---- extract diagonals to LDS ----
    // diag(r,r): r<8 -> VGPR r, lane r ; r>=8 -> VGPR r-8, lane r+16
    const bool dval = (lane < 8) || (lane >= 24);
    const int  didx = (lane < 8) ? lane : (lane - 24);
    const int  drow = (lane < 8) ? lane : (lane - 16);
    if (dval) {
        float* rw = &red[wave][0][0];
        rw[ 0 * 16 + drow] = vsel8(acc_gg,   didx);
        rw[ 1 * 16 + drow] = vsel8(acc_ng1,  didx);
        rw[ 2 * 16 + drow] = vsel8(acc_ng2,  didx);
        rw[ 3 * 16 + drow] = vsel8(acc_ss,   didx);
        rw[ 4 * 16 + drow] = vsel8(acc_ns1,  didx);
        rw[ 5 * 16 + drow] = vsel8(acc_ns2,  didx);
        rw[ 6 * 16 + drow] = vsel8(acc_cc,   didx);
        rw[ 7 * 16 + drow] = vsel8(acc_nga,  didx);
        rw[ 8 * 16 + drow] = vsel8(acc_nsa,  didx);
        rw[ 9 * 16 + drow] = vsel8(acc_att,  didx);
        rw[10 * 16 + drow] = vsel8(acc_gn,   didx);
        rw[11 * 16 + drow] = vsel8(acc_ng2n, didx);
        rw[12 * 16 + drow] = vsel8(acc_sn,   didx);
        rw[13 * 16 + drow] = vsel8(acc_ns2n, didx);
        rw[14 * 16 + drow] = vsel8(acc_cn,   didx);
        rw[15 * 16 + drow] = vsel8(acc_nsan, didx);
    }

    // ---- per-row finish: cosines, InfoNCE softplus terms, attention gate ----
    const float EPS  = 1e-8f;
    const float invT = 10.0f;  // 1 / 0.1
    if (lane < 16) {
        const float* rr = &red[wave][0][0];
        float dgg  = rr[ 0 * 16 + lane], ng1v = rr[ 1 * 16 + lane], ng2v = rr[ 2 * 16 + lane];
        float dss  = rr[ 3 * 16 + lane], ns1v = rr[ 4 * 16 + lane], ns2v = rr[ 5 * 16 + lane];
        float dcc  = rr[ 6 * 16 + lane], ngav = rr[ 7 * 16 + lane], nsav = rr[ 8 * 16 + lane];
        float attv = rr[ 9 * 16 + lane];
        float dgn  = rr[10 * 16 + lane], ng2n = rr[11 * 16 + lane];
        float dsn  = rr[12 * 16 + lane], ns2n = rr[13 * 16 + lane];
        float dcn  = rr[14 * 16 + lane], nsan = rr[15 * 16 + lane];

        float sg1 = sqrtf(ng1v), sg2 = sqrtf(ng2v);
        float ss1 = sqrtf(ns1v), ss2 = sqrtf(ns2v);
        float sga = sqrtf(ngav), ssa = sqrtf(nsav);

        float pos_g = dgg / fmaxf(sg1 * sg2, EPS);
        float neg_g = dgn / fmaxf(sg1 * sqrtf(ng2n), EPS);
        float pos_s = dss / fmaxf(ss1 * ss2, EPS);
        float neg_s = dsn / fmaxf(ss1 * sqrtf(ns2n), EPS);
        float pos_c = dcc / fmaxf(sga * ssa, EPS);
        float neg_c = dcn / fmaxf(sga * sqrtf(nsan), EPS);

        float li = softplus_f((neg_g - pos_g) * invT)
                 + softplus_f((neg_s - pos_s) * invT)
                 + softplus_f((neg_c - pos_c) * invT);
        float alpha = 1.f / (1.f + expf(-attv));

        // reuse slots 0/1 for broadcast (reads above already consumed)
        red[wave][0][lane] = alpha;
        red[wave][1][lane] = li;
    }

    if (lane == 0) {
        float s = 0.f;
        for (int r = 0; r < 16; ++r) s += red[wave][1][r];
        atomicAdd(loss, s / (float)n_items);
    }

    // ---- output rows: alpha*ga + (1-alpha)*sa, coalesced float4/lane ----
    for (int r = 0; r < 16; ++r) {
        float al = red[wave][0][r];
        int base = (r0 + r) * 128 + lane * 4;
        v4f vg1 = *(const v4f*)(g1 + base);
        v4f vg2 = *(const v4f*)(g2 + base);
        v4f vs1 = *(const v4f*)(s1 + base);
        v4f vs2 = *(const v4f*)(s2 + base);
        v4f ga = 0.5f * (vg1 + vg2);
        v4f sa = 0.5f * (vs1 + vs2);
        v4f o  = al * ga + (1.f - al) * sa;
        *(v4f*)(out + base) = o;
    }
}

extern "C" void kernel_launch(void* const* d_in, const int* in_sizes, int n_in,
                              void* d_out, int out_size, void* d_ws, size_t ws_size,
                              hipStream_t stream) {
    const float* g1   = (const float*)d_in[0];
    const float* g2   = (const float*)d_in[1];
    const float* s1   = (const float*)d_in[2];
    const float* s2   = (const float*)d_in[3];
    const float* attw = (const float*)d_in[4];
    const int*   ng   = (const int*)d_in[5];
    const int*   ns   = (const int*)d_in[6];
    const int*   nc   = (const int*)d_in[7];

    float* out = (float*)d_out;
    const int n_items = in_sizes[0] / 128;       // 200000
    float* loss = out + (size_t)n_items * 128;   // last output element

    zero_scalar_kernel<<<1, 1, 0, stream>>>(loss);

    const int n_tiles = (n_items + 15) / 16;     // 12500 (exact)
    const int blocks  = (n_tiles + 3) / 4;       // 4 waves (tiles) per 128-thread block
    mvrec_wmma_kernel<<<blocks, 128, 0, stream>>>(g1, g2, s1, s2, attw, ng, ns, nc,
                                                  out, loss, n_items, n_tiles);
}